// EventSegmentationNetwork_89902255439913
// MI455X (gfx1250) — compile-verified
//
#include <hip/hip_runtime.h>

// ---------------------------------------------------------------------------
// EventSegmentationNetwork (Mamba-style selective scan) for gfx1250 / MI455X
// fp32 end-to-end; GEMMs use V_WMMA_F32_16X16X4_F32 (exact fp32 matrix core)
// with 32x64 register blocking per wave (2 M-tiles x 4 N-tiles).
// Scan is a 3-pass chunked parallel linear recurrence.
// ---------------------------------------------------------------------------

#define NTOK   65536
#define DMODEL 256
#define DINNER 512
#define DSTATE 16
#define DCONV  4
#define CHUNK  128
#define NCHUNK (NTOK / CHUNK)   // 512

typedef float v2f __attribute__((ext_vector_type(2)));
typedef float v8f __attribute__((ext_vector_type(8)));

enum { EPI_NONE = 0, EPI_SIGMOID = 1, EPI_SOFTPLUS = 2 };

// ---------------------------------------------------------------------------
// Y = X @ W^T (+bias, +activation) using WMMA f32 16x16x4, register-blocked.
// X: (M x K) row-major, ld = ldx.  W: (Ncols x K) row-major, ld = ldw.
// One wave computes a (16*MT) x (16*NT) macro-tile: MT*NT accumulators,
// per K-step MT+NT fragment loads feed MT*NT WMMAs (A/B reuse in registers).
// A frag (16x4): lane<16 -> M=lane, K={k,k+1}; lane>=16 -> K={k+2,k+3}
// B frag (4x16): lane<16 -> N=lane, K={k,k+1}; lane>=16 -> K={k+2,k+3}
// C/D: vgpr r, lane<16 -> (M=r, N=lane); lane>=16 -> (M=r+8, N=lane-16)
// ---------------------------------------------------------------------------
template <int EPI, int MT, int NT>
__launch_bounds__(256)
__global__ void gemm_wmma_blk(const float* __restrict__ X, int ldx,
                              const float* __restrict__ W, int ldw,
                              const float* __restrict__ bias,
                              float* __restrict__ Y, int ldy,
                              int M, int Ncols, int K) {
    const int lane = threadIdx.x & 31;
    const int wid  = threadIdx.x >> 5;
    const int lo   = lane & 15;
    const int hi   = lane >> 4;
    const int ntn  = Ncols / (16 * NT);
    const int gw   = blockIdx.x * (blockDim.x >> 5) + wid;
    const int tM   = gw / ntn;
    const int tN   = gw % ntn;
    if (tM * 16 * MT >= M) return;

    const float* Xr[MT];
    const float* Wr[NT];
#pragma unroll
    for (int i = 0; i < MT; ++i)
        Xr[i] = X + (size_t)(tM * 16 * MT + i * 16 + lo) * ldx + 2 * hi;
#pragma unroll
    for (int j = 0; j < NT; ++j)
        Wr[j] = W + (size_t)(tN * 16 * NT + j * 16 + lo) * ldw + 2 * hi;

    v8f acc[MT][NT] = {};
#pragma unroll 4
    for (int k = 0; k < K; k += 4) {
        v2f a[MT], b[NT];
#pragma unroll
        for (int i = 0; i < MT; ++i) a[i] = *(const v2f*)(Xr[i] + k);
#pragma unroll
        for (int j = 0; j < NT; ++j) b[j] = *(const v2f*)(Wr[j] + k);
#pragma unroll
        for (int i = 0; i < MT; ++i)
#pragma unroll
            for (int j = 0; j < NT; ++j)
                acc[i][j] = __builtin_amdgcn_wmma_f32_16x16x4_f32(
                    /*neg_a=*/false, a[i], /*neg_b=*/false, b[j],
                    /*c_mod=*/(short)0, acc[i][j],
                    /*reuse_a=*/false, /*reuse_b=*/false);
    }

#pragma unroll
    for (int j = 0; j < NT; ++j) {
        const int col = tN * 16 * NT + j * 16 + lo;
        const float bv = bias ? bias[col] : 0.0f;
#pragma unroll
        for (int i = 0; i < MT; ++i) {
#pragma unroll
            for (int r = 0; r < 8; ++r) {
                const int row = tM * 16 * MT + i * 16 + r + 8 * hi;
                float v = acc[i][j][r] + bv;
                if (EPI == EPI_SIGMOID)  v = 1.0f / (1.0f + __expf(-v));
                if (EPI == EPI_SOFTPLUS) v = (v > 20.0f) ? v : log1pf(__expf(v));
                Y[(size_t)row * ldy + col] = v;
            }
        }
    }
}

// ---------------------------------------------------------------------------
// Depthwise causal conv (width 4) over the token dim + bias + SiLU.
// x_in = xz[:, 0:512].
// ---------------------------------------------------------------------------
__global__ void conv_silu_kernel(const float* __restrict__ xz,
                                 const float* __restrict__ cw,
                                 const float* __restrict__ cb,
                                 float* __restrict__ xact) {
    const size_t idx = (size_t)blockIdx.x * blockDim.x + threadIdx.x;
    const int c = (int)(idx & (DINNER - 1));
    const long n = (long)(idx >> 9);
    float acc = cb[c];
#pragma unroll
    for (int k = 0; k < DCONV; ++k) {
        const long m = n - (DCONV - 1) + k;
        if (m >= 0) acc += cw[c * DCONV + k] * xz[(size_t)m * (2 * DINNER) + c];
    }
    xact[idx] = acc / (1.0f + __expf(-acc));   // SiLU
}

// ---------------------------------------------------------------------------
// Guidance head: g = gelu(layernorm(guidance @ gg1_w^T + gg1_b)).
// One wave per token row; 16 channels per lane; shuffle reduction (wave32).
// ---------------------------------------------------------------------------
__global__ void guidance_kernel(const float* __restrict__ guid,
                                const float* __restrict__ w,
                                const float* __restrict__ b,
                                const float* __restrict__ lng,
                                const float* __restrict__ lnb,
                                float* __restrict__ gout) {
    const int lane = threadIdx.x & 31;
    const int wid  = threadIdx.x >> 5;
    const int row  = blockIdx.x * 8 + wid;
    const float g0 = guid[row * 3 + 0];
    const float g1 = guid[row * 3 + 1];
    const float g2 = guid[row * 3 + 2];
    float v[16];
    float sum = 0.0f, sq = 0.0f;
#pragma unroll
    for (int i = 0; i < 16; ++i) {
        const int c = lane + 32 * i;
        const float t = g0 * w[c * 3 + 0] + g1 * w[c * 3 + 1] + g2 * w[c * 3 + 2] + b[c];
        v[i] = t; sum += t; sq += t * t;
    }
#pragma unroll
    for (int m = 16; m >= 1; m >>= 1) {
        sum += __shfl_xor(sum, m, 32);
        sq  += __shfl_xor(sq,  m, 32);
    }
    const float mu  = sum * (1.0f / DINNER);
    const float var = sq * (1.0f / DINNER) - mu * mu;
    const float rs  = rsqrtf(var + 1e-5f);
#pragma unroll
    for (int i = 0; i < 16; ++i) {
        const int c = lane + 32 * i;
        const float t = (v[i] - mu) * rs * lng[c] + lnb[c];
        gout[(size_t)row * DINNER + c] = 0.5f * t * (1.0f + erff(t * 0.70710678118654752f));
    }
}

// x_mod = x_act * sigmoid-gate(g_input)   (gates already passed sigmoid epilogue)
__global__ void xmod_kernel(const float* __restrict__ xact,
                            const float* __restrict__ gates,
                            float* __restrict__ xmod) {
    const size_t idx = (size_t)blockIdx.x * blockDim.x + threadIdx.x;
    const int c = (int)(idx & (DINNER - 1));
    const size_t n = idx >> 9;
    xmod[idx] = xact[idx] * gates[n * (3 * DINNER) + c];
}

// ---------------------------------------------------------------------------
// Selective scan, 3-pass chunked linear recurrence.
// Pass 1: per-chunk local state (h starting from 0) + per-channel sum(delta).
// Chunk decay product is exp(A * sum(delta)) so only the sum is needed.
// ---------------------------------------------------------------------------
__launch_bounds__(DINNER)
__global__ void scan_pass1(const float* __restrict__ delta,
                           const float* __restrict__ u,
                           const float* __restrict__ xdbl,   // (N,32): B|C
                           const float* __restrict__ A_log,
                           float* __restrict__ Sdelta,       // (NCHUNK, DINNER)
                           float* __restrict__ Hloc) {       // (NCHUNK, DINNER, 16)
    __shared__ float bc[CHUNK * 32];
    const int j = blockIdx.x;
    const int d = threadIdx.x;
    for (int i = threadIdx.x; i < CHUNK * 32; i += DINNER)
        bc[i] = xdbl[(size_t)j * CHUNK * 32 + i];
    __syncthreads();

    float A[16], h[16];
#pragma unroll
    for (int s = 0; s < 16; ++s) { A[s] = -__expf(A_log[d * 16 + s]); h[s] = 0.0f; }
    float sd = 0.0f;
    const size_t base = (size_t)j * CHUNK * DINNER + d;
    for (int t = 0; t < CHUNK; ++t) {
        const float dl = delta[base + (size_t)t * DINNER];
        const float uu = u[base + (size_t)t * DINNER];
        sd += dl;
        const float du = dl * uu;
        const float* B = &bc[t * 32];
#pragma unroll
        for (int s = 0; s < 16; ++s)
            h[s] = __expf(dl * A[s]) * h[s] + du * B[s];
    }
    Sdelta[(size_t)j * DINNER + d] = sd;
    const size_t hb = ((size_t)j * DINNER + d) * 16;
#pragma unroll
    for (int s = 0; s < 16; ++s) Hloc[hb + s] = h[s];
}

// Pass 2: inter-chunk recurrence. One thread per (channel, state) pair.
__global__ void scan_pass2(const float* __restrict__ A_log,
                           const float* __restrict__ Sdelta,
                           const float* __restrict__ Hloc,
                           float* __restrict__ Hin) {
    const int g = blockIdx.x * blockDim.x + threadIdx.x;   // 0..8191
    const int d = g >> 4;
    const int s = g & 15;
    const float A = -__expf(A_log[d * 16 + s]);
    float carry = 0.0f;
    for (int j = 0; j < NCHUNK; ++j) {
        const size_t idx = ((size_t)j * DINNER + d) * 16 + s;
        Hin[idx] = carry;
        carry = __expf(A * Sdelta[(size_t)j * DINNER + d]) * carry + Hloc[idx];
    }
}

// Pass 3: replay with incoming state; emit y * silu(z) * g_output.
__launch_bounds__(DINNER)
__global__ void scan_pass3(const float* __restrict__ delta,
                           const float* __restrict__ u,
                           const float* __restrict__ xdbl,
                           const float* __restrict__ A_log,
                           const float* __restrict__ Dvec,
                           const float* __restrict__ Hin,
                           const float* __restrict__ xz,     // z = xz[:,512:1024]
                           const float* __restrict__ gates,  // g_out = gates[:,1024:1536]
                           float* __restrict__ ymod) {
    __shared__ float bc[CHUNK * 32];
    const int j = blockIdx.x;
    const int d = threadIdx.x;
    for (int i = threadIdx.x; i < CHUNK * 32; i += DINNER)
        bc[i] = xdbl[(size_t)j * CHUNK * 32 + i];
    __syncthreads();

    float A[16], h[16];
    const size_t hb = ((size_t)j * DINNER + d) * 16;
#pragma unroll
    for (int s = 0; s < 16; ++s) { A[s] = -__expf(A_log[d * 16 + s]); h[s] = Hin[hb + s]; }
    const float Dp = Dvec[d];

    for (int t = 0; t < CHUNK; ++t) {
        const size_t n = (size_t)j * CHUNK + t;
        const float dl = delta[n * DINNER + d];
        const float uu = u[n * DINNER + d];
        const float du = dl * uu;
        float y = Dp * uu;
        const float* B = &bc[t * 32];
#pragma unroll
        for (int s = 0; s < 16; ++s) {
            h[s] = __expf(dl * A[s]) * h[s] + du * B[s];
            y += h[s] * B[16 + s];   // C part
        }
        const float z  = xz[n * (2 * DINNER) + DINNER + d];
        const float sz = z / (1.0f + __expf(-z));
        const float go = gates[n * (3 * DINNER) + 2 * DINNER + d];
        ymod[n * DINNER + d] = y * sz * go;
    }
}

// ---------------------------------------------------------------------------
extern "C" void kernel_launch(void* const* d_in, const int* in_sizes, int n_in,
                              void* d_out, int out_size, void* d_ws, size_t ws_size,
                              hipStream_t stream) {
    (void)in_sizes; (void)n_in; (void)out_size; (void)ws_size;
    const float* x         = (const float*)d_in[0];
    const float* guidance  = (const float*)d_in[1];
    const float* in_proj_w = (const float*)d_in[2];
    const float* conv_w    = (const float*)d_in[3];
    const float* conv_b    = (const float*)d_in[4];
    const float* x_proj_w  = (const float*)d_in[5];
    const float* dt_proj_w = (const float*)d_in[6];
    const float* dt_proj_b = (const float*)d_in[7];
    const float* gg1_w     = (const float*)d_in[8];
    const float* gg1_b     = (const float*)d_in[9];
    const float* ln_g      = (const float*)d_in[10];
    const float* ln_b      = (const float*)d_in[11];
    const float* gg2_w     = (const float*)d_in[12];
    const float* gg2_b     = (const float*)d_in[13];
    const float* A_log     = (const float*)d_in[14];
    const float* Dvec      = (const float*)d_in[15];
    const float* out_w     = (const float*)d_in[16];
    float* out = (float*)d_out;
    float* ws  = (float*)d_ws;

    const size_t Nt = NTOK;
    // workspace layout (floats); x_act region recycled for y, g region for delta
    float* xz     = ws;                         // N x 1024
    float* x_act  = xz    + Nt * 1024;          // N x 512   (later: y_mod)
    float* gbuf   = x_act + Nt * 512;           // N x 512   (later: delta)
    float* gates  = gbuf  + Nt * 512;           // N x 1536
    float* x_mod  = gates + Nt * 1536;          // N x 512
    float* x_dbl  = x_mod + Nt * 512;           // N x 32
    float* Sdelta = x_dbl + Nt * 32;            // 512 x 512
    float* Hloc   = Sdelta + (size_t)NCHUNK * DINNER;        // 512 x 512 x 16
    float* Hin    = Hloc   + (size_t)NCHUNK * DINNER * 16;   // 512 x 512 x 16
    float* deltaB = gbuf;       // alias after gbuf is dead
    float* y_mod  = x_act;      // alias after x_act is dead

    const int TPB = 256;
    // waves per block = 8; macro-tile = 32 rows x (16*NT) cols
    // 1) xz = x @ in_proj_w^T   (M=N, K=256, Ncols=1024)
    gemm_wmma_blk<EPI_NONE, 2, 4><<<(NTOK/32)*(1024/64)/8, TPB, 0, stream>>>(
        x, DMODEL, in_proj_w, DMODEL, nullptr, xz, 2*DINNER, NTOK, 2*DINNER, DMODEL);
    // 2) causal depthwise conv + SiLU
    conv_silu_kernel<<<(NTOK*DINNER)/TPB, TPB, 0, stream>>>(xz, conv_w, conv_b, x_act);
    // 3a) guidance -> LN -> GELU
    guidance_kernel<<<NTOK/8, TPB, 0, stream>>>(guidance, gg1_w, gg1_b, ln_g, ln_b, gbuf);
    // 3b) gates = sigmoid(g @ gg2_w^T + gg2_b)   (K=512, Ncols=1536)
    gemm_wmma_blk<EPI_SIGMOID, 2, 4><<<(NTOK/32)*(1536/64)/8, TPB, 0, stream>>>(
        gbuf, DINNER, gg2_w, DINNER, gg2_b, gates, 3*DINNER, NTOK, 3*DINNER, DINNER);
    // 3c) x_mod = x_act * g_input
    xmod_kernel<<<(NTOK*DINNER)/TPB, TPB, 0, stream>>>(x_act, gates, x_mod);
    // 4a) delta = softplus(x_mod @ dt_proj_w^T + dt_proj_b)  (overwrites gbuf)
    gemm_wmma_blk<EPI_SOFTPLUS, 2, 4><<<(NTOK/32)*(DINNER/64)/8, TPB, 0, stream>>>(
        x_mod, DINNER, dt_proj_w, DINNER, dt_proj_b, deltaB, DINNER, NTOK, DINNER, DINNER);
    // 4b) x_dbl = x_mod @ x_proj_w^T  (Ncols=32: B|C) -- 32x32 macro-tiles
    gemm_wmma_blk<EPI_NONE, 2, 2><<<(NTOK/32)*(32/32)/8, TPB, 0, stream>>>(
        x_mod, DINNER, x_proj_w, DINNER, nullptr, x_dbl, 2*DSTATE, NTOK, 2*DSTATE, DINNER);
    // 5) chunked parallel selective scan
    scan_pass1<<<NCHUNK, DINNER, 0, stream>>>(deltaB, x_mod, x_dbl, A_log, Sdelta, Hloc);
    scan_pass2<<<(DINNER*DSTATE)/TPB, TPB, 0, stream>>>(A_log, Sdelta, Hloc, Hin);
    scan_pass3<<<NCHUNK, DINNER, 0, stream>>>(deltaB, x_mod, x_dbl, A_log, Dvec, Hin,
                                              xz, gates, y_mod);
    // 6) out = y_mod @ out_proj_w^T  (Ncols=256)
    gemm_wmma_blk<EPI_NONE, 2, 4><<<(NTOK/32)*(DMODEL/64)/8, TPB, 0, stream>>>(
        y_mod, DINNER, out_w, DINNER, nullptr, out, DMODEL, NTOK, DMODEL, DINNER);
}